// VectorQuantizer_71227737637023
// MI455X (gfx1250) — compile-verified
//
#include <hip/hip_runtime.h>

typedef float v2f __attribute__((ext_vector_type(2)));
typedef float v8f __attribute__((ext_vector_type(8)));

#define N_CODES     16384
#define EMBED_DIM   256
#define NPIX        16384      // 16 * 32 * 32
#define PIX_PER_IMG 1024       // 32 * 32
#define LDS_STRIDE  260        // 256 + 4 pad -> conflict-free f32 banks
#define TILE_CODES  32         // codes per iteration (two 16-wide N tiles)
#define NT          (N_CODES / TILE_CODES)  // 512 iterations

// ---------------------------------------------------------------------------
// Kernel 1: per-code squared norms ||e_k||^2 -> d_ws; also init loss scalar.
// ---------------------------------------------------------------------------
__global__ __launch_bounds__(256) void vq_norms_kernel(
    const float* __restrict__ emb, float* __restrict__ norms,
    float* __restrict__ loss_out) {
  const int wave = threadIdx.x >> 5;
  const int lane = threadIdx.x & 31;
  const int code = blockIdx.x * 8 + wave;
  const float* row = emb + (size_t)code * EMBED_DIM;
  float s = 0.0f;
#pragma unroll
  for (int j = 0; j < EMBED_DIM / 32; ++j) {
    float v = row[lane + 32 * j];
    s += v * v;
  }
#pragma unroll
  for (int off = 16; off >= 1; off >>= 1) s += __shfl_down(s, off, 32);
  if (lane == 0) norms[code] = s;
  if (blockIdx.x == 0 && threadIdx.x == 0) loss_out[0] = 0.0f;
}

// ---------------------------------------------------------------------------
// Kernel 2: fused (-2 z.e^T + ||e||^2) argmin via V_WMMA_F32_16X16X4_F32.
//   - 1 wave  = one 16-pixel M tile (A resident: 64 x v2f = 128 VGPRs)
//   - 1 block = 8 waves; 32-code B tiles staged in LDS by
//     GLOBAL_LOAD_ASYNC_TO_LDS_B128, triple-buffered, 1 barrier/iter
//   - 2 independent WMMA accumulator chains per iteration for ILP
// ---------------------------------------------------------------------------
__global__ __launch_bounds__(256) void vq_argmin_kernel(
    const float* __restrict__ z, const float* __restrict__ emb,
    const float* __restrict__ norms, float* __restrict__ zq,
    float* __restrict__ idx_out) {
  __shared__ __align__(16) float ldsB[3][TILE_CODES * LDS_STRIDE];
  __shared__ int ldsIdx[8][16];

  const int tid  = threadIdx.x;
  const int wave = tid >> 5;
  const int lane = tid & 31;
  const int half = lane >> 4;   // which 16-lane half
  const int l15  = lane & 15;

  const int mtile = blockIdx.x * 8 + wave;  // 1024 M tiles total
  const int m0    = mtile * 16;

  // ---- async-copy helper state: this thread's slice of a 32x256 tile ------
  // thread t owns code row (t>>3), 128B chunk ((t&7)*32 floats): 8 x b128
  const int arow = tid >> 3;          // 0..31
  const int acol = (tid & 7) * 32;    // float offset of 128B chunk
  const unsigned lbase[3] = {
      (unsigned)(uintptr_t)&ldsB[0][arow * LDS_STRIDE + acol],
      (unsigned)(uintptr_t)&ldsB[1][arow * LDS_STRIDE + acol],
      (unsigned)(uintptr_t)&ldsB[2][arow * LDS_STRIDE + acol]};
  const uint64_t gbase =
      (uint64_t)(uintptr_t)(emb + (size_t)arow * EMBED_DIM + acol);

#define ASYNC_B128(LADDR, GADDR, OFF)                                          \
  asm volatile("global_load_async_to_lds_b128 %0, %1, off offset:" #OFF        \
               :: "v"(LADDR), "v"(GADDR) : "memory")
#define ISSUE_TILE(KB, BUF)                                                    \
  do {                                                                         \
    unsigned la = lbase[BUF];                                                  \
    uint64_t ga = gbase + (uint64_t)(KB) * (EMBED_DIM * 4);                    \
    ASYNC_B128(la, ga, 0);   ASYNC_B128(la, ga, 16);                           \
    ASYNC_B128(la, ga, 32);  ASYNC_B128(la, ga, 48);                           \
    ASYNC_B128(la, ga, 64);  ASYNC_B128(la, ga, 80);                           \
    ASYNC_B128(la, ga, 96);  ASYNC_B128(la, ga, 112);                          \
  } while (0)

  // ---- load resident A fragments: 64 k-steps of 16x4 f32 ------------------
  // ISA layout: lane = M (l&15); VGPR0 = K 2*half, VGPR1 = K 2*half+1
  v2f afrag[64];
  {
    const int g   = m0 + l15;     // global pixel row; tile stays in one image
    const int b   = g >> 10;
    const int pix = g & 1023;
    const float* zb = z + (size_t)b * EMBED_DIM * PIX_PER_IMG + pix;
#pragma unroll
    for (int k = 0; k < 64; ++k) {
      const int c = 4 * k + 2 * half;
      afrag[k].x = zb[(size_t)c * PIX_PER_IMG];
      afrag[k].y = zb[(size_t)(c + 1) * PIX_PER_IMG];
    }
  }

  float minv[8];
  int   mini[8];
#pragma unroll
  for (int r = 0; r < 8; ++r) { minv[r] = 3.4e38f; mini[r] = 0; }

  ISSUE_TILE(0, 0);  // prologue: tile 0 -> buffer 0

  for (int kt = 0; kt < NT; ++kt) {
    const int kb  = kt * TILE_CODES;
    const int buf = kt % 3;
    if (kt + 1 < NT) {
      ISSUE_TILE(kb + TILE_CODES, (kt + 1) % 3);  // overlap next-tile copy
      asm volatile("s_wait_asynccnt 0x8" ::: "memory");  // tile kt arrived
    } else {
      asm volatile("s_wait_asynccnt 0x0" ::: "memory");
    }
    __syncthreads();  // everyone's tile-kt data visible; (kt+1)%3 buf is free

    const float* Bb = ldsB[buf];
    v8f acc0 = {0.f, 0.f, 0.f, 0.f, 0.f, 0.f, 0.f, 0.f};
    v8f acc1 = {0.f, 0.f, 0.f, 0.f, 0.f, 0.f, 0.f, 0.f};
#pragma unroll
    for (int k = 0; k < 64; ++k) {
      // B layout: lane = N (l&15); VGPR0 = row K=2*half, VGPR1 = K=2*half+1
      v2f b0 = *(const v2f*)&Bb[l15 * LDS_STRIDE + 4 * k + 2 * half];
      v2f b1 = *(const v2f*)&Bb[(16 + l15) * LDS_STRIDE + 4 * k + 2 * half];
      acc0 = __builtin_amdgcn_wmma_f32_16x16x4_f32(
          false, afrag[k], false, b0, (short)0, acc0, false, false);
      acc1 = __builtin_amdgcn_wmma_f32_16x16x4_f32(
          false, afrag[k], false, b1, (short)0, acc1, false, false);
    }

    // D layout: VGPR r -> M = r + 8*half, N = l15
    const float n0 = norms[kb + l15];
    const float n1 = norms[kb + 16 + l15];
    const int   c0 = kb + l15;
    const int   c1 = kb + 16 + l15;
#pragma unroll
    for (int r = 0; r < 8; ++r) {
      const float d0 = n0 - 2.0f * acc0[r];
      if (d0 < minv[r]) { minv[r] = d0; mini[r] = c0; }
      const float d1 = n1 - 2.0f * acc1[r];
      if (d1 < minv[r]) { minv[r] = d1; mini[r] = c1; }
    }
  }

  // ---- cross-lane argmin over the 16 lanes of each half (xor stays in half)
#pragma unroll
  for (int r = 0; r < 8; ++r) {
#pragma unroll
    for (int off = 8; off >= 1; off >>= 1) {
      const float ov = __shfl_xor(minv[r], off, 32);
      const int   oi = __shfl_xor(mini[r], off, 32);
      if (ov < minv[r] || (ov == minv[r] && oi < mini[r])) {
        minv[r] = ov; mini[r] = oi;
      }
    }
  }
  if (l15 == 0) {
#pragma unroll
    for (int r = 0; r < 8; ++r) {
      const int m = r + 8 * half;      // lane0 -> M 0..7, lane16 -> M 8..15
      ldsIdx[wave][m] = mini[r];
      idx_out[m0 + m] = (float)mini[r];
    }
  }
  __syncthreads();

  // ---- gather z_q (NCHW): lanes 0-15 cover 16 consecutive pixels (64B runs)
  {
    const int b    = m0 >> 10;
    const int pix0 = m0 & 1023;
    const int myIdx = ldsIdx[wave][l15];          // code for my pixel
    const float* erow = emb + (size_t)myIdx * EMBED_DIM;
    float* zqb = zq + (size_t)b * EMBED_DIM * PIX_PER_IMG + pix0 + l15;
#pragma unroll 4
    for (int cc = 0; cc < 128; ++cc) {
      const int ch = 2 * cc + half;
      zqb[(size_t)ch * PIX_PER_IMG] = erow[ch];
    }
  }
#undef ASYNC_B128
#undef ISSUE_TILE
}

// ---------------------------------------------------------------------------
extern "C" void kernel_launch(void* const* d_in, const int* in_sizes, int n_in,
                              void* d_out, int out_size, void* d_ws,
                              size_t ws_size, hipStream_t stream) {
  const float* z   = (const float*)d_in[0];  // (16,256,32,32) f32
  const float* emb = (const float*)d_in[1];  // (16384,256) f32

  float* out     = (float*)d_out;
  float* zq      = out;                                         // 4,194,304
  float* loss    = out + (size_t)16 * EMBED_DIM * PIX_PER_IMG;  // 1
  float* idx_out = loss + 1;                                    // 16,384
  float* norms   = (float*)d_ws;                                // 16,384

  vq_norms_kernel<<<N_CODES / 8, 256, 0, stream>>>(emb, norms, loss);
  vq_argmin_kernel<<<NPIX / (16 * 8), 256, 0, stream>>>(z, emb, norms, zq,
                                                        idx_out);
}